// StreamMix_54382875902584
// MI455X (gfx1250) — compile-verified
//
#include <hip/hip_runtime.h>
#include <hip/hip_bf16.h>

typedef float v2f __attribute__((ext_vector_type(2)));
typedef float v8f __attribute__((ext_vector_type(8)));

#define N 4096
#define SK_EPS 1e-12f
#define SK_ITERS 20

// ---------------------------------------------------------------------------
// A = exp(logits), vectorized float4.  One HBM read + one HBM write of 64 MB;
// A then stays resident in the 192 MB L2 for all 40 matvec passes.
// ---------------------------------------------------------------------------
__global__ __launch_bounds__(256) void sk_exp_kernel(const float* __restrict__ L,
                                                     float* __restrict__ A) {
    size_t i = (size_t)blockIdx.x * blockDim.x + threadIdx.x;  // float4 index
    const float4* l4 = (const float4*)L;
    float4* a4 = (float4*)A;
    float4 x = l4[i];
    float4 y;
    y.x = expf(x.x);
    y.y = expf(x.y);
    y.z = expf(x.z);
    y.w = expf(x.w);
    a4[i] = y;
}

// u = v = 1
__global__ __launch_bounds__(256) void sk_init_uv(float* __restrict__ u,
                                                  float* __restrict__ v) {
    int i = blockIdx.x * 256 + threadIdx.x;
    u[i] = 1.0f;
    v[i] = 1.0f;
}

// ---------------------------------------------------------------------------
// Row pass: rowsum_i = u_i * (A v)_i ; u_i <- u_i / max(rowsum_i, eps).
// One 256-thread block per row; coalesced float4 loads; LDS tree reduce.
// ---------------------------------------------------------------------------
__global__ __launch_bounds__(256) void sk_row_update(const float* __restrict__ A,
                                                     float* __restrict__ u,
                                                     const float* __restrict__ v) {
    const int row = blockIdx.x;
    const int tid = threadIdx.x;
    const float4* ar = (const float4*)(A + (size_t)row * N);
    const float4* v4 = (const float4*)v;

    float s = 0.0f;
#pragma unroll
    for (int k = 0; k < 4; ++k) {
        int j = tid + k * 256;  // float4 index within row (1024 per row)
        float4 a = ar[j];
        float4 b = v4[j];
        s = fmaf(a.x, b.x, fmaf(a.y, b.y, fmaf(a.z, b.z, fmaf(a.w, b.w, s))));
    }

    __shared__ float red[256];
    red[tid] = s;
    __syncthreads();
    for (int off = 128; off > 0; off >>= 1) {
        if (tid < off) red[tid] += red[tid + off];
        __syncthreads();
    }
    if (tid == 0) {
        float ui = u[row];
        float rs = ui * red[0];              // actual row sum of diag(u) A diag(v)
        u[row] = ui / fmaxf(rs, SK_EPS);
    }
}

// ---------------------------------------------------------------------------
// Column pass via V_WMMA_F32_16X16X4_F32:
//   D = W(16x4) x B(4x16), with W_mk = u[k] replicated over m, B = 4x16 tile
//   of A (contiguous 16-float row segments).  Every row of D then holds the
//   partial weighted column sums  sum_k u_k * A[k][j].
// Layouts per CDNA5 ISA (32-bit 16x4 A-operand): VGPR0 = {K0 | K2} across the
// two lane halves, VGPR1 = {K1 | K3}; B operand filled with the mirrored
// mapping so k-pairing matches.
// Block = 8 waves; wave w covers rows [w*512, w*512+512) of a 16-column tile;
// cross-wave reduce in LDS, then threads 0..15 update v.
// ---------------------------------------------------------------------------
__global__ __launch_bounds__(256) void sk_col_update_wmma(const float* __restrict__ A,
                                                          const float* __restrict__ u,
                                                          float* __restrict__ v) {
    const int tid  = threadIdx.x;
    const int lane = tid & 31;
    const int wave = tid >> 5;
    const int half = lane >> 4;          // 0: lanes 0-15, 1: lanes 16-31
    const int jb   = blockIdx.x * 16;    // 16-column tile
    const int col  = jb + (lane & 15);

    v8f acc = {};
    const int rbeg = wave * (N / 8);
    const int rend = rbeg + (N / 8);
    for (int k = rbeg; k < rend; k += 4) {
        const int r0 = k + half * 2;     // this half-wave holds K = r0, r0+1
        v2f wa, wb;
        wa[0] = u[r0];                   // W_m,K0/K2  (broadcast within half)
        wa[1] = u[r0 + 1];               // W_m,K1/K3
        wb[0] = A[(size_t)r0 * N + col];       // B_K0/K2, n
        wb[1] = A[(size_t)(r0 + 1) * N + col]; // B_K1/K3, n
        acc = __builtin_amdgcn_wmma_f32_16x16x4_f32(
            /*neg_a=*/false, wa, /*neg_b=*/false, wb,
            /*c_mod=*/(short)0, acc, /*reuse_a=*/false, /*reuse_b=*/false);
    }

    // D rows are identical; take M=0 row: VGPR0, lanes 0-15 (N = 0..15).
    __shared__ float part[8][16];
    if (lane < 16) part[wave][lane] = acc[0];
    __syncthreads();

    if (tid < 16) {
        float cs = 0.0f;
#pragma unroll
        for (int w = 0; w < 8; ++w) cs += part[w][tid];
        float vj = v[jb + tid];
        float t = vj * cs;               // actual column sum of diag(u) A diag(v)
        v[jb + tid] = vj / fmaxf(t, SK_EPS);
    }
}

// ---------------------------------------------------------------------------
// Final: out_ij = u_i * A_ij * v_j, in place on d_out (float4 vectorized).
// ---------------------------------------------------------------------------
__global__ __launch_bounds__(256) void sk_final_scale(float* __restrict__ A,
                                                      const float* __restrict__ u,
                                                      const float* __restrict__ v) {
    size_t i = (size_t)blockIdx.x * blockDim.x + threadIdx.x;  // float4 index
    int row = (int)(i >> 10);     // 1024 float4 per row
    int c4  = (int)(i & 1023);
    float4* a4 = (float4*)A;
    const float4* v4 = (const float4*)v;
    float4 a = a4[i];
    float4 vv = v4[c4];
    float ui = u[row];
    a.x *= ui * vv.x;
    a.y *= ui * vv.y;
    a.z *= ui * vv.z;
    a.w *= ui * vv.w;
    a4[i] = a;
}

extern "C" void kernel_launch(void* const* d_in, const int* in_sizes, int n_in,
                              void* d_out, int out_size, void* d_ws, size_t ws_size,
                              hipStream_t stream) {
    (void)in_sizes; (void)n_in; (void)out_size; (void)ws_size;

    const float* logits = (const float*)d_in[0];
    float* A = (float*)d_out;            // 4096x4096 f32: A = exp(logits), scaled in place at end
    float* u = (float*)d_ws;             // 4096 f32
    float* v = u + N;                    // 4096 f32

    const int vec_blocks = (N * N / 4) / 256;  // 16384

    sk_exp_kernel<<<vec_blocks, 256, 0, stream>>>(logits, A);
    sk_init_uv<<<N / 256, 256, 0, stream>>>(u, v);

    for (int it = 0; it < SK_ITERS; ++it) {
        sk_row_update<<<N, 256, 0, stream>>>(A, u, v);
        sk_col_update_wmma<<<N / 16, 256, 0, stream>>>(A, u, v);
    }

    sk_final_scale<<<vec_blocks, 256, 0, stream>>>(A, u, v);
}